// SimpleNet_28518582846260
// MI455X (gfx1250) — compile-verified
//
#include <hip/hip_runtime.h>

typedef __attribute__((ext_vector_type(4))) float v4f;
typedef __attribute__((ext_vector_type(2))) float v2f;
typedef __attribute__((ext_vector_type(8))) float v8f;

#define D_FEAT 1024
#define WAVES_PER_BLOCK 8

// out[row] = sum_d tanh(x[row][d]*w[d] + b[d])^2
// One wave (32 lanes) per row. Streaming f32, NT hint on x (no reuse, 268MB).
__global__ __launch_bounds__(WAVES_PER_BLOCK * 32)
void affine_tanh_sqsum_kernel(const float* __restrict__ x,
                              const float* __restrict__ w,
                              const float* __restrict__ b,
                              float* __restrict__ out) {
    const int lane = threadIdx.x & 31;
    const int wave = threadIdx.x >> 5;
    const int row  = blockIdx.x * WAVES_PER_BLOCK + wave;

    const v4f* __restrict__ xr = (const v4f*)(x + (size_t)row * D_FEAT);
    const v4f* __restrict__ w4 = (const v4f*)w;
    const v4f* __restrict__ b4 = (const v4f*)b;

    float s = 0.0f;
#pragma unroll
    for (int i = 0; i < D_FEAT / (32 * 4); ++i) {   // 8 iterations
        const int idx = i * 32 + lane;              // coalesced float4 index
        v4f xv = __builtin_nontemporal_load(&xr[idx]);  // streaming: NT hint
        v4f wv = w4[idx];                               // cached (8KB working set)
        v4f bv = b4[idx];
#pragma unroll
        for (int j = 0; j < 4; ++j) {
            float t = __builtin_fmaf(xv[j], wv[j], bv[j]);
            float h;
#if __has_builtin(__builtin_amdgcn_tanhf)
            h = __builtin_amdgcn_tanhf(t);          // v_tanh_f32 (TRANS32 on gfx1250)
#else
            h = tanhf(t);
#endif
            s = __builtin_fmaf(h, h, s);
        }
    }

    // Cross-lane reduction of the 32 per-lane partials.
#if __has_builtin(__builtin_amdgcn_wmma_f32_16x16x4_f32)
    // A (16x4 f32, 2 VGPRs): lanes 0-15 hold M=lane, K={0,1}; lanes 16-31 K={2,3}.
    // Put partial in K0/K2 slot, zero in K1/K3. B = all-ones (layout-invariant).
    // => D[m][n] = s_m + s_(m+16) for every n.
    v2f a;    a[0] = s;    a[1] = 0.0f;
    v2f ones; ones[0] = 1.0f; ones[1] = 1.0f;
    v8f c = {};
    c = __builtin_amdgcn_wmma_f32_16x16x4_f32(
        /*neg_a=*/false, a, /*neg_b=*/false, ones,
        /*c_mod=*/(short)0, c, /*reuse_a=*/false, /*reuse_b=*/false);
    // lane<16:  sum_j c[j] = (s_0..7) + (s_16..23)
    // lane>=16: sum_j c[j] = (s_8..15) + (s_24..31)
    float t = ((c[0] + c[1]) + (c[2] + c[3])) + ((c[4] + c[5]) + (c[6] + c[7]));
    float total = t + __shfl_xor(t, 16, 32);
#else
    float total = s;
#pragma unroll
    for (int off = 16; off > 0; off >>= 1) total += __shfl_xor(total, off, 32);
#endif

    if (lane == 0) out[row] = total;
}

extern "C" void kernel_launch(void* const* d_in, const int* in_sizes, int n_in,
                              void* d_out, int out_size, void* d_ws, size_t ws_size,
                              hipStream_t stream) {
    const float* x = (const float*)d_in[0];   // [B, 1024] f32
    const float* w = (const float*)d_in[1];   // [1024]    f32
    const float* b = (const float*)d_in[2];   // [1024]    f32
    float* out = (float*)d_out;               // [B, 1]    f32

    const int rows = in_sizes[0] / D_FEAT;    // 65536
    const int blocks = rows / WAVES_PER_BLOCK;
    affine_tanh_sqsum_kernel<<<blocks, WAVES_PER_BLOCK * 32, 0, stream>>>(x, w, b, out);
}